// PointNetFeaturePropagation_55954833933049
// MI455X (gfx1250) — compile-verified
//
#include <hip/hip_runtime.h>
#include <hip/hip_bf16.h>

typedef __attribute__((ext_vector_type(2))) float v2f;
typedef __attribute__((ext_vector_type(8))) float v8f;
typedef __attribute__((ext_vector_type(4))) unsigned int v4u;
typedef __attribute__((ext_vector_type(4))) int v4i;
typedef __attribute__((ext_vector_type(8))) int v8i;

#define B_      16
#define N_      4096
#define S_      1024
#define D1_     64
#define D2_     256
#define IN_CH_  320
#define OUT_CH_ 256
#define XPAD_   40   // LDS row stride in dwords: 32 data + 8 pad (2*40 mod 64 = 16 -> conflict-free halves)

#if __has_builtin(__builtin_amdgcn_tensor_load_to_lds)
#define HAVE_TDM 1
#endif

// ---------------------------------------------------------------------------
// Kernel 1: three_nn. One thread per query point; xyz2 (+ squared norms)
// staged in LDS. Expanded |a|^2+|b|^2-2ab form matches the reference algebra.
// ---------------------------------------------------------------------------
__global__ __launch_bounds__(256) void three_nn_kernel(
    const float* __restrict__ xyz1, const float* __restrict__ xyz2,
    int* __restrict__ idx_out, float* __restrict__ w_out)
{
    __shared__ float sx[S_], sy[S_], sz[S_], sn[S_];
    const int b = blockIdx.y;
    const int n = blockIdx.x * 256 + threadIdx.x;

    const float* p2 = xyz2 + (size_t)b * S_ * 3;
    for (int s = threadIdx.x; s < S_; s += 256) {
        float x = p2[s * 3 + 0], y = p2[s * 3 + 1], z = p2[s * 3 + 2];
        sx[s] = x; sy[s] = y; sz[s] = z; sn[s] = x * x + y * y + z * z;
    }
    __syncthreads();

    const float* q = xyz1 + ((size_t)b * N_ + n) * 3;
    const float qx = q[0], qy = q[1], qz = q[2];
    const float qn = qx * qx + qy * qy + qz * qz;

    float d0 = 3.4e38f, d1 = 3.4e38f, d2 = 3.4e38f;
    int   i0 = 0, i1 = 0, i2 = 0;
    for (int s = 0; s < S_; ++s) {
        float dot = qx * sx[s] + qy * sy[s] + qz * sz[s];
        float d   = (qn + sn[s]) - 2.0f * dot;
        if (d < d0)      { d2 = d1; i2 = i1; d1 = d0; i1 = i0; d0 = d; i0 = s; }
        else if (d < d1) { d2 = d1; i2 = i1; d1 = d;  i1 = s; }
        else if (d < d2) { d2 = d;  i2 = s; }
    }
    const float r0 = 1.0f / (d0 + 1e-8f);
    const float r1 = 1.0f / (d1 + 1e-8f);
    const float r2 = 1.0f / (d2 + 1e-8f);
    const float inv = 1.0f / (r0 + r1 + r2);

    const size_t o = ((size_t)b * N_ + n) * 3;
    idx_out[o + 0] = i0; idx_out[o + 1] = i1; idx_out[o + 2] = i2;
    w_out[o + 0] = r0 * inv; w_out[o + 1] = r1 * inv; w_out[o + 2] = r2 * inv;
}

// ---------------------------------------------------------------------------
// Kernel 2: three_interpolate. Block = (batch, 64 points); idx/weights staged
// in LDS; each block emits 256 channels x 64 points, coalesced in n.
// ---------------------------------------------------------------------------
__global__ __launch_bounds__(256) void interp_kernel(
    const float* __restrict__ points2, const int* __restrict__ idx,
    const float* __restrict__ wgt, float* __restrict__ interp)
{
    __shared__ int   sI[64 * 3];
    __shared__ float sW[64 * 3];
    const int b  = blockIdx.y;
    const int n0 = blockIdx.x * 64;

    const size_t base = ((size_t)b * N_ + n0) * 3;
    for (int t = threadIdx.x; t < 192; t += 256) {
        sI[t] = idx[base + t];
        sW[t] = wgt[base + t];
    }
    __syncthreads();

    const float* p2  = points2 + (size_t)b * D2_ * S_;
    float*       out = interp  + (size_t)b * D2_ * N_;
    #pragma unroll 4
    for (int i = 0; i < 64; ++i) {
        const int lin = i * 256 + threadIdx.x;
        const int c   = lin >> 6;
        const int nn  = lin & 63;
        const float* row = p2 + (size_t)c * S_;
        const float v = sW[nn * 3 + 0] * row[sI[nn * 3 + 0]]
                      + sW[nn * 3 + 1] * row[sI[nn * 3 + 1]]
                      + sW[nn * 3 + 2] * row[sI[nn * 3 + 2]];
        out[(size_t)c * N_ + n0 + nn] = v;
    }
}

#ifdef HAVE_TDM
// Build a 2D-tile D# (tile: 32 contiguous elems x `rows` rows, row stride
// `rowstride` elems, 4B data) and issue TENSOR_LOAD_TO_LDS. pad_enable with
// pad_interval=32dw / pad_amount=8dw gives an LDS row stride of XPAD_ dwords.
__device__ __forceinline__ void tdm_load_panel(const float* gsrc,
                                               unsigned int lds_byte_off,
                                               unsigned int rows,
                                               unsigned int rowstride)
{
    const unsigned long long ga = (unsigned long long)(uintptr_t)gsrc;
    v4u g0;
    g0[0] = 1u;                                   // count=1, user descriptor
    g0[1] = lds_byte_off;                         // lds_addr (bytes)
    g0[2] = (unsigned int)ga;                     // global_addr[31:0]
    g0[3] = (unsigned int)((ga >> 32) & 0x01FFFFFFu) | (2u << 30); // addr[56:32] | type=2

    v8i g1;
    g1[0] = (int)((2u << 16) | (1u << 20) | (4u << 22) | (7u << 25)); // 4B elems, pad: every 32dw add 8dw
    g1[1] = (int)((rowstride & 0xFFFFu) << 16);                       // tensor_dim0[15:0]
    g1[2] = (int)(((rowstride >> 16) & 0xFFFFu) | ((rows & 0xFFFFu) << 16)); // dim0[31:16] | dim1[15:0]
    g1[3] = (int)(((rows >> 16) & 0xFFFFu) | (32u << 16));            // dim1[31:16] | tile_dim0=32
    g1[4] = (int)(rows & 0xFFFFu);                                    // tile_dim1=rows, tile_dim2=0
    g1[5] = (int)rowstride;                                           // tensor_dim0_stride[31:0]
    g1[6] = 0;                                                        // stride[47:32] | dim1_stride[15:0]
    g1[7] = 0;
    v4i z4 = {0, 0, 0, 0};
#if defined(__clang_major__) && __clang_major__ >= 23
    v8i z8 = {0, 0, 0, 0, 0, 0, 0, 0};
    __builtin_amdgcn_tensor_load_to_lds(g0, g1, z4, z4, z8, 0);
#else
    __builtin_amdgcn_tensor_load_to_lds(g0, g1, z4, z4, 0);
#endif
}
#endif

// ---------------------------------------------------------------------------
// Kernel 3: fused concat + 1x1 conv via V_WMMA_F32_16X16X4_F32.
// Block = (batch, 32-point panel). x-panel (320ch x 32pt) is staged once in
// LDS (TDM DMA when available), row stride XPAD_=40 dwords -> the two lane
// halves of a WMMA B-read hit disjoint LDS bank groups. Each of the 8 waves
// computes a 2(m) x 2(n) block of 16x16 tiles: per K-step 2 VMEM + 4 DS + 4
// WMMA, so the wave is matrix-issue limited rather than load-issue limited.
// ---------------------------------------------------------------------------
__global__ __launch_bounds__(256) void conv_wmma_kernel(
    const float* __restrict__ Wm, const float* __restrict__ points1,
    const float* __restrict__ interp, float* __restrict__ y)
{
    __shared__ float sX[IN_CH_ * XPAD_];   // 320 * 40 * 4B = 50 KB of 320 KB WGP LDS

    const int n0 = blockIdx.x * 32;
    const int b  = blockIdx.y;

#ifdef HAVE_TDM
    if (threadIdx.x < 32) {   // one wave issues both DMAs (TDM issues once per wave)
        const unsigned int lds0 = (unsigned int)(uintptr_t)&sX[0];
        tdm_load_panel(points1 + (size_t)b * D1_ * N_ + n0, lds0, D1_, N_);
        tdm_load_panel(interp  + (size_t)b * D2_ * N_ + n0,
                       lds0 + (unsigned int)(D1_ * XPAD_ * 4), D2_, N_);
        __builtin_amdgcn_s_wait_tensorcnt(0);
    }
    __syncthreads();
#else
    {
        const float* s1 = points1 + (size_t)b * D1_ * N_ + n0;
        const float* s2 = interp  + (size_t)b * D2_ * N_ + n0;
        for (int t = threadIdx.x; t < IN_CH_ * 32; t += 256) {
            const int c = t >> 5, nn = t & 31;
            const float v = (c < D1_) ? s1[(size_t)c * N_ + nn]
                                      : s2[(size_t)(c - D1_) * N_ + nn];
            sX[c * XPAD_ + nn] = v;
        }
        __syncthreads();
    }
#endif

    const int wave = threadIdx.x >> 5;
    const int lane = threadIdx.x & 31;
    const int half = lane >> 4;
    const int l    = lane & 15;
    const int m0   = wave << 5;            // 2 m-tiles per wave -> 8 waves cover 256 ch

    const float* Wrow0 = Wm + (size_t)(m0 + l) * IN_CH_;
    const float* Wrow1 = Wm + (size_t)(m0 + 16 + l) * IN_CH_;

    v8f a00 = {}, a01 = {}, a10 = {}, a11 = {};
    #pragma unroll 2
    for (int k = 0; k < IN_CH_; k += 4) {
        // A: 16x4 f32 tile; lanes 0-15 carry K=0,1; lanes 16-31 carry K=2,3
        v2f wa0 = ((const v2f*)(Wrow0 + k))[half];
        v2f wa1 = ((const v2f*)(Wrow1 + k))[half];
        // B: mirrored K split, columns from the LDS panel (conflict-free)
        const int c0 = k + 2 * half;
        const float* r0 = &sX[c0 * XPAD_];
        const float* r1 = &sX[(c0 + 1) * XPAD_];
        v2f b0, b1;
        b0.x = r0[l];      b0.y = r1[l];        // n-tile 0
        b1.x = r0[16 + l]; b1.y = r1[16 + l];   // n-tile 1
        a00 = __builtin_amdgcn_wmma_f32_16x16x4_f32(false, wa0, false, b0, (short)0, a00, false, false);
        a01 = __builtin_amdgcn_wmma_f32_16x16x4_f32(false, wa0, false, b1, (short)0, a01, false, false);
        a10 = __builtin_amdgcn_wmma_f32_16x16x4_f32(false, wa1, false, b0, (short)0, a10, false, false);
        a11 = __builtin_amdgcn_wmma_f32_16x16x4_f32(false, wa1, false, b1, (short)0, a11, false, false);
    }

    // D layout: VGPR r, lanes 0-15 -> M=r, lanes 16-31 -> M=r+8
    float* yb = y + (size_t)b * OUT_CH_ * N_;
    #pragma unroll
    for (int r = 0; r < 8; ++r) {
        const int ch0 = m0 + 8 * half + r;
        const int ch1 = m0 + 16 + 8 * half + r;
        yb[(size_t)ch0 * N_ + n0 + l]      = a00[r];
        yb[(size_t)ch0 * N_ + n0 + 16 + l] = a01[r];
        yb[(size_t)ch1 * N_ + n0 + l]      = a10[r];
        yb[(size_t)ch1 * N_ + n0 + 16 + l] = a11[r];
    }
}

// ---------------------------------------------------------------------------
// Kernel 4: per-channel batch stats (biased, training mode). One block per
// channel; deterministic LDS tree reduction (no atomics).
// ---------------------------------------------------------------------------
__global__ __launch_bounds__(256) void bn_stats_kernel(
    const float* __restrict__ y, float* __restrict__ mean_out,
    float* __restrict__ rstd_out)
{
    const int c = blockIdx.x;
    float s = 0.0f, sq = 0.0f;
    for (int b = 0; b < B_; ++b) {
        const float* row = y + ((size_t)b * OUT_CH_ + c) * N_;
        for (int n = threadIdx.x; n < N_; n += 256) {
            const float v = row[n];
            s += v; sq += v * v;
        }
    }
    __shared__ float ss[256], ssq[256];
    ss[threadIdx.x] = s; ssq[threadIdx.x] = sq;
    __syncthreads();
    for (int off = 128; off > 0; off >>= 1) {
        if (threadIdx.x < off) {
            ss[threadIdx.x]  += ss[threadIdx.x + off];
            ssq[threadIdx.x] += ssq[threadIdx.x + off];
        }
        __syncthreads();
    }
    if (threadIdx.x == 0) {
        const float inv = 1.0f / (float)(B_ * N_);
        const float m   = ss[0] * inv;
        const float var = ssq[0] * inv - m * m;
        mean_out[c] = m;
        rstd_out[c] = rsqrtf(var + 1e-5f);
    }
}

// ---------------------------------------------------------------------------
// Kernel 5: normalize + affine + ReLU, in place on d_out, float4 vectorized.
// ---------------------------------------------------------------------------
__global__ __launch_bounds__(256) void bn_apply_kernel(
    float* __restrict__ y, const float* __restrict__ mean,
    const float* __restrict__ rstd, const float* __restrict__ gamma,
    const float* __restrict__ beta)
{
    const size_t e = (size_t)blockIdx.x * 256 + threadIdx.x;   // float4 index
    const int c = (int)((e / (N_ / 4)) % OUT_CH_);
    const float sc = rstd[c] * gamma[c];
    const float sh = beta[c] - mean[c] * sc;
    float4* p = (float4*)y + e;
    float4 v = *p;
    v.x = fmaxf(v.x * sc + sh, 0.0f);
    v.y = fmaxf(v.y * sc + sh, 0.0f);
    v.z = fmaxf(v.z * sc + sh, 0.0f);
    v.w = fmaxf(v.w * sc + sh, 0.0f);
    *p = v;
}

// ---------------------------------------------------------------------------
extern "C" void kernel_launch(void* const* d_in, const int* in_sizes, int n_in,
                              void* d_out, int out_size, void* d_ws, size_t ws_size,
                              hipStream_t stream)
{
    const float* xyz1    = (const float*)d_in[0];   // [B,N,3]
    const float* xyz2    = (const float*)d_in[1];   // [B,S,3]
    const float* points1 = (const float*)d_in[2];   // [B,64,N]
    const float* points2 = (const float*)d_in[3];   // [B,256,S]
    const float* Wm      = (const float*)d_in[4];   // [256,320]
    const float* gamma   = (const float*)d_in[5];   // [256]
    const float* beta    = (const float*)d_in[6];   // [256]
    float* y = (float*)d_out;                       // [B,256,N] (y, then final)

    // workspace layout
    char* ws = (char*)d_ws;
    int*   idx    = (int*)  (ws);                                    // 786432 B
    float* wgt    = (float*)(ws + (786432));                         // 786432 B
    float* interp = (float*)(ws + (2 * 786432));                     // 64 MB
    float* meanb  = (float*)(ws + (2 * 786432) + (size_t)67108864);  // 1 KB
    float* rstdb  = meanb + 256;

    // 1) three_nn: grid (N/256, B)
    three_nn_kernel<<<dim3(N_ / 256, B_), 256, 0, stream>>>(xyz1, xyz2, idx, wgt);

    // 2) interpolate: grid (N/64, B)
    interp_kernel<<<dim3(N_ / 64, B_), 256, 0, stream>>>(points2, idx, wgt, interp);

    // 3) concat + 1x1 conv via fp32 WMMA + TDM-staged LDS panels
    conv_wmma_kernel<<<dim3(N_ / 32, B_), 256, 0, stream>>>(Wm, points1, interp, y);

    // 4) BN batch stats: one block per channel
    bn_stats_kernel<<<dim3(OUT_CH_), 256, 0, stream>>>(y, meanb, rstdb);

    // 5) normalize + ReLU in place (float4 granularity)
    bn_apply_kernel<<<dim3((size_t)B_ * OUT_CH_ * N_ / 4 / 256), 256, 0, stream>>>(
        y, meanb, rstdb, gamma, beta);
}